// PromptModule_65412351918558
// MI455X (gfx1250) — compile-verified
//
#include <hip/hip_runtime.h>
#include <math.h>

typedef __attribute__((ext_vector_type(2))) float v2f;
typedef __attribute__((ext_vector_type(4))) float v4f;
typedef __attribute__((ext_vector_type(8))) float v8f;

#define BS      64
#define SEQ     2048
#define D       768
#define POOL    64
#define PLEN    5
#define TOPK    5
#define OUTSEQ  (TOPK * PLEN + SEQ)          /* 2073 */
#define OUTROW  ((size_t)OUTSEQ * D)         /* 1,592,064 floats per batch */
#define XROW    ((size_t)SEQ * D)            /* 1,572,864 floats per batch */
#define PROMPT_FLOATS (TOPK * PLEN * D)      /* 19,200 */
#define LOSS_OFF ((size_t)BS * OUTROW)       /* 101,892,096 */

// ---------------------------------------------------------------------------
// 1) L2-normalize: 128 blocks (0..63 = cls_feature rows, 64..127 = prompt_key)
// ---------------------------------------------------------------------------
__global__ __launch_bounds__(256) void normalize_kernel(
    const float* __restrict__ cls, const float* __restrict__ pkey,
    float* __restrict__ qn, float* __restrict__ kn) {
  __shared__ float red[8];
  const int row    = blockIdx.x & 63;
  const bool iskey = blockIdx.x >= 64;
  const float* src = (iskey ? pkey : cls) + (size_t)row * D;
  float*       dst = (iskey ? kn : qn) + (size_t)row * D;

  float v[3], sq = 0.f;
#pragma unroll
  for (int i = 0; i < 3; ++i) {
    v[i] = src[threadIdx.x + i * 256];
    sq += v[i] * v[i];
  }
  // wave32 reduce
#pragma unroll
  for (int off = 16; off > 0; off >>= 1) sq += __shfl_down(sq, off, 32);
  if ((threadIdx.x & 31) == 0) red[threadIdx.x >> 5] = sq;
  __syncthreads();
  if (threadIdx.x == 0) {
    float t = 0.f;
#pragma unroll
    for (int i = 0; i < 8; ++i) t += red[i];
    red[0] = rsqrtf(fmaxf(t, 1e-12f));
  }
  __syncthreads();
  const float scale = red[0];
#pragma unroll
  for (int i = 0; i < 3; ++i) dst[threadIdx.x + i * 256] = v[i] * scale;
}

// ---------------------------------------------------------------------------
// 2) sim = qn @ kn^T via V_WMMA_F32_16X16X4_F32.
//    One block of 512 threads = 16 waves; wave w owns 16x16 tile (w>>2, w&3).
// ---------------------------------------------------------------------------
__global__ __launch_bounds__(512) void sim_wmma_kernel(
    const float* __restrict__ qn, const float* __restrict__ kn,
    float* __restrict__ sim) {
  const int wave = threadIdx.x >> 5;   // 0..15
  const int lane = threadIdx.x & 31;
  const int tm   = wave >> 2;          // row tile (batch)
  const int tn   = wave & 3;           // col tile (pool)
  const int half = lane >> 4;          // 0: K={0,1}; 1: K={2,3}
  const int l16  = lane & 15;

  // A frag source: query row (M), B frag source: key row (N), same lane pattern
  const float* qrow = qn + (size_t)(tm * 16 + l16) * D + half * 2;
  const float* krow = kn + (size_t)(tn * 16 + l16) * D + half * 2;

  v8f acc = {};
#pragma unroll 8
  for (int k0 = 0; k0 < D; k0 += 4) {
    v2f a = *(const v2f*)(qrow + k0);
    v2f b = *(const v2f*)(krow + k0);
    acc = __builtin_amdgcn_wmma_f32_16x16x4_f32(
        /*neg_a=*/false, a, /*neg_b=*/false, b,
        /*c_mod=*/(short)0, acc, /*reuse_a=*/false, /*reuse_b=*/false);
  }
  // C/D layout: VGPR r -> M = 16*tm + r + 8*half, N = 16*tn + l16
  const int n = tn * 16 + l16;
#pragma unroll
  for (int r = 0; r < 8; ++r) {
    const int m = tm * 16 + r + half * 8;
    sim[m * POOL + n] = acc[r];
  }
}

// ---------------------------------------------------------------------------
// 3) top-5 per batch row + gather selected prompts into the output head.
//    64 blocks (one per batch), 256 threads.
// ---------------------------------------------------------------------------
__global__ __launch_bounds__(256) void topk_gather_kernel(
    const float* __restrict__ sim, const float* __restrict__ prompt,
    float* __restrict__ out) {
  __shared__ int s_idx[TOPK];
  const int b = blockIdx.x;
  if (threadIdx.x == 0) {
    const float* s = sim + b * POOL;
    unsigned long long chosen = 0ull;
    for (int k = 0; k < TOPK; ++k) {
      float best = -INFINITY;
      int bi = 0;
      for (int i = 0; i < POOL; ++i) {
        const bool taken = (chosen >> i) & 1ull;
        if (!taken && s[i] > best) { best = s[i]; bi = i; }  // ties -> lowest idx
      }
      chosen |= 1ull << bi;
      s_idx[k] = bi;
    }
  }
  __syncthreads();

  v4f* dst = (v4f*)(out + (size_t)b * OUTROW);
  const int n4 = PROMPT_FLOATS / 4;        // 4800 float4 per batch
  for (int j = threadIdx.x; j < n4; j += 256) {
    const int slot = j / (D / 4);          // 0..24
    const int c4   = j - slot * (D / 4);
    const int k    = slot / PLEN;
    const int l    = slot - k * PLEN;
    const v4f* src = (const v4f*)(prompt + ((size_t)s_idx[k] * PLEN + l) * D);
    dst[(size_t)slot * (D / 4) + c4] = src[c4];
  }
}

// ---------------------------------------------------------------------------
// 4) loss = sum(kn * qn) / BS  (single block reduction, deterministic)
// ---------------------------------------------------------------------------
__global__ __launch_bounds__(256) void loss_kernel(
    const float* __restrict__ qn, const float* __restrict__ kn,
    float* __restrict__ out_loss) {
  __shared__ float red[8];
  float s = 0.f;
  for (int i = threadIdx.x; i < BS * D; i += 256) s += qn[i] * kn[i];
#pragma unroll
  for (int off = 16; off > 0; off >>= 1) s += __shfl_down(s, off, 32);
  if ((threadIdx.x & 31) == 0) red[threadIdx.x >> 5] = s;
  __syncthreads();
  if (threadIdx.x == 0) {
    float t = 0.f;
#pragma unroll
    for (int i = 0; i < 8; ++i) t += red[i];
    *out_loss = t / (float)BS;
  }
}

// ---------------------------------------------------------------------------
// 5) Streaming copy of x_embed into out tail. Bandwidth-dominant: ~811 MB of
//    traffic -> non-temporal b128 load/store to bypass cache retention.
//    Grid: (XROW/4/(256*4), BS) = (384, 64); 4 float4 per thread.
// ---------------------------------------------------------------------------
__global__ __launch_bounds__(256) void copy_x_kernel(
    const v4f* __restrict__ x, float* __restrict__ out) {
  const int b = blockIdx.y;
  const v4f* src = x + (size_t)b * (XROW / 4);
  v4f*       dst = (v4f*)(out + (size_t)b * OUTROW + PROMPT_FLOATS);
  const size_t base = (size_t)blockIdx.x * (256 * 4) + threadIdx.x;
#pragma unroll
  for (int u = 0; u < 4; ++u) {
    const size_t i = base + (size_t)u * 256;
    v4f v = __builtin_nontemporal_load(src + i);
    __builtin_nontemporal_store(v, dst + i);
  }
}

// ---------------------------------------------------------------------------
extern "C" void kernel_launch(void* const* d_in, const int* in_sizes, int n_in,
                              void* d_out, int out_size, void* d_ws, size_t ws_size,
                              hipStream_t stream) {
  (void)in_sizes; (void)n_in; (void)out_size; (void)ws_size;
  const float* x_embed = (const float*)d_in[0];  // (64, 2048, 768)
  const float* cls     = (const float*)d_in[1];  // (64, 768)
  const float* prompt  = (const float*)d_in[2];  // (64, 5, 768)
  const float* pkey    = (const float*)d_in[3];  // (64, 768)
  float* out = (float*)d_out;

  float* ws  = (float*)d_ws;
  float* qn  = ws;                   // 64*768
  float* kn  = ws + BS * D;          // 64*768
  float* sim = ws + 2 * BS * D;      // 64*64

  normalize_kernel<<<dim3(128), dim3(256), 0, stream>>>(cls, pkey, qn, kn);
  sim_wmma_kernel<<<dim3(1), dim3(512), 0, stream>>>(qn, kn, sim);
  topk_gather_kernel<<<dim3(BS), dim3(256), 0, stream>>>(sim, prompt, out);
  loss_kernel<<<dim3(1), dim3(256), 0, stream>>>(qn, kn, out + LOSS_OFF);
  copy_x_kernel<<<dim3((unsigned)(XROW / 4 / (256 * 4)), BS), dim3(256), 0, stream>>>(
      (const v4f*)x_embed, out);
}